// MultiHeadAttentionOpCUDA_79972291052181
// MI455X (gfx1250) — compile-verified
//
#include <hip/hip_runtime.h>
#include <hip/hip_bf16.h>
#include <math.h>

// ---------------------------------------------------------------------------
// MultiHeadAttentionOp (feature-axis low-rank attention), MI455X / gfx1250.
//
// Math:  scores[b,h,i,j] = 0.5 * x_i * x_j * G[h,i,j],  G = Qe @ Ke^T (rank 4)
//        head_out[b,h,i] = softmax_j(scores) . (x_j * w[h,j]),  w = Ve . out_proj
//        out = x + sigmoid(gate) * sum_h softmax(head_weights)_h * head_out
//
// Kernel 1: G via V_WMMA_F32_16X16X4_F32 (exact rank-4 shape match).
// Kernel 2: w[h,j] projection (tiny).
// Kernel 3: fused per-(b,h) row softmax — exp-bound; G streamed from L2 with
//           global_load_b128; x_j / u_j held in VGPRs across the row loop.
// Kernel 4: head combine + gated residual.
// ---------------------------------------------------------------------------

#define D_FEAT   256
#define N_HEADS  32
#define RANK     4
#define BATCH_N  64

typedef __attribute__((ext_vector_type(2))) float v2f;
typedef __attribute__((ext_vector_type(8))) float v8f;

// ---------------- Kernel 1: Gram tensor G[h,i,j] via WMMA -------------------
// One wave handles a 16-row strip (h, ti): loads A (16x4 of Qe) once, then
// 16 WMMAs sweeping tj, reusing A. 512 waves total.
__global__ void __launch_bounds__(256)
mha_gram_wmma_kernel(const float* __restrict__ Qe,
                     const float* __restrict__ Ke,
                     float* __restrict__ G) {
    const int lane = threadIdx.x & 31;
    const int wave = threadIdx.x >> 5;
    const int task = blockIdx.x * 8 + wave;     // 0 .. 511
    const int h  = task >> 4;
    const int ti = task & 15;

    const int half = lane >> 4;                 // 0: K={0,1}, 1: K={2,3}
    const int l16  = lane & 15;

    // A matrix (16x4 f32): lanes 0-15 rows M=0..15 with K=0/1, lanes 16-31 K=2/3
    const int i = ti * 16 + l16;
    const float* qa = Qe + ((size_t)(h * D_FEAT + i)) * RANK + 2 * half;
    v2f a; a.x = qa[0]; a.y = qa[1];

    float* gbase = G + ((size_t)h * D_FEAT + (size_t)ti * 16) * D_FEAT;

#pragma unroll
    for (int tj = 0; tj < 16; ++tj) {
        // B matrix (4x16 f32) = Ke^T tile: same half/K striping, N = l16
        const int j = tj * 16 + l16;
        const float* kb = Ke + ((size_t)(h * D_FEAT + j)) * RANK + 2 * half;
        v2f bm; bm.x = kb[0]; bm.y = kb[1];

        v8f c = {};
        c = __builtin_amdgcn_wmma_f32_16x16x4_f32(
                /*neg_a=*/false, a, /*neg_b=*/false, bm,
                /*c_mod=*/(short)0, c, /*reuse_a=*/false, /*reuse_b=*/false);

        // D layout: VGPR v -> row M=v (lanes 0-15) / M=v+8 (lanes 16-31), N=l16
#pragma unroll
        for (int v = 0; v < 8; ++v) {
            gbase[(size_t)(v + 8 * half) * D_FEAT + tj * 16 + l16] = c[v];
        }
    }
}

// ---------------- Kernel 2: w[h,j] = Ve[h,j,:] . out_proj[h,:] --------------
__global__ void __launch_bounds__(256)
mha_vproj_kernel(const float* __restrict__ Ve,
                 const float* __restrict__ op,
                 float* __restrict__ w) {
    const int idx = blockIdx.x * 256 + threadIdx.x;     // h*256 + j
    const int h = idx >> 8;
    const float* v = Ve + (size_t)idx * RANK;
    const float* o = op + h * RANK;
    w[idx] = v[0] * o[0] + v[1] * o[1] + v[2] * o[2] + v[3] * o[3];
}

// ---------------- Kernel 3: fused row softmax + matvec ----------------------
// Block = one (b,h) pair, 256 threads = 8 waves, each wave owns 32 rows.
// Lane owns j in {4L..4L+3, 128+4L..128+4L+3}; x_j, u_j live in VGPRs across
// the whole row loop; per row: 2x global_load_b128 of G + 1 LDS broadcast.
__global__ void __launch_bounds__(256)
mha_rows_kernel(const float* __restrict__ x,
                const float* __restrict__ G,
                const float* __restrict__ w,
                float* __restrict__ hout) {
    __shared__ __align__(16) float sx[D_FEAT];
    __shared__ __align__(16) float su[D_FEAT];

    const int h = blockIdx.x & (N_HEADS - 1);
    const int b = blockIdx.x >> 5;
    const int t = threadIdx.x;

    const float xv = x[b * D_FEAT + t];
    sx[t] = xv;
    su[t] = xv * w[h * D_FEAT + t];
    __syncthreads();

    const int lane = t & 31;
    const int wave = t >> 5;
    const int j0 = lane * 4;            // lane's first float4 group
    const int j1 = 128 + lane * 4;      // lane's second float4 group

    // Row-invariant operands: keep in VGPRs for all 32 rows.
    const float4 xj0 = *(const float4*)(sx + j0);
    const float4 xj1 = *(const float4*)(sx + j1);
    const float4 uj0 = *(const float4*)(su + j0);
    const float4 uj1 = *(const float4*)(su + j1);

    const float* Gh   = G + (size_t)h * D_FEAT * D_FEAT;
    float*       hrow = hout + ((size_t)(b * N_HEADS + h)) * D_FEAT;

    for (int ii = 0; ii < 32; ++ii) {
        const int i = wave * 32 + ii;
        const float ci = 0.5f * sx[i];             // LDS broadcast read
        const float* grow = Gh + (size_t)i * D_FEAT;
        __builtin_prefetch(grow + D_FEAT, 0, 1);   // global_prefetch_b8 next row

        const float4 g0 = *(const float4*)(grow + j0);   // global_load_b128
        const float4 g1 = *(const float4*)(grow + j1);   // global_load_b128

        float s[8];
        s[0] = ci * (xj0.x * g0.x);
        s[1] = ci * (xj0.y * g0.y);
        s[2] = ci * (xj0.z * g0.z);
        s[3] = ci * (xj0.w * g0.w);
        s[4] = ci * (xj1.x * g1.x);
        s[5] = ci * (xj1.y * g1.y);
        s[6] = ci * (xj1.z * g1.z);
        s[7] = ci * (xj1.w * g1.w);

        float m = fmaxf(fmaxf(fmaxf(s[0], s[1]), fmaxf(s[2], s[3])),
                        fmaxf(fmaxf(s[4], s[5]), fmaxf(s[6], s[7])));
#pragma unroll
        for (int off = 16; off > 0; off >>= 1)
            m = fmaxf(m, __shfl_xor(m, off, 32));

        float p0 = __expf(s[0] - m);
        float p1 = __expf(s[1] - m);
        float p2 = __expf(s[2] - m);
        float p3 = __expf(s[3] - m);
        float p4 = __expf(s[4] - m);
        float p5 = __expf(s[5] - m);
        float p6 = __expf(s[6] - m);
        float p7 = __expf(s[7] - m);

        float l   = ((p0 + p1) + (p2 + p3)) + ((p4 + p5) + (p6 + p7));
        float acc = ((p0 * uj0.x + p1 * uj0.y) + (p2 * uj0.z + p3 * uj0.w)) +
                    ((p4 * uj1.x + p5 * uj1.y) + (p6 * uj1.z + p7 * uj1.w));
#pragma unroll
        for (int off = 16; off > 0; off >>= 1) {
            l   += __shfl_xor(l,   off, 32);
            acc += __shfl_xor(acc, off, 32);
        }
        if (lane == 0) hrow[i] = acc / l;
    }
}

// ---------------- Kernel 4: head softmax combine + gated residual -----------
__global__ void __launch_bounds__(256)
mha_combine_kernel(const float* __restrict__ x,
                   const float* __restrict__ hout,
                   const float* __restrict__ hwts,
                   const float* __restrict__ gate_logit,
                   float* __restrict__ out) {
    const int idx = blockIdx.x * 256 + threadIdx.x;    // b*256 + f
    const int b = idx >> 8;
    const int f = idx & (D_FEAT - 1);

    float hm = -3.402823466e38f;
#pragma unroll
    for (int h = 0; h < N_HEADS; ++h) hm = fmaxf(hm, hwts[h]);

    float hsum = 0.0f, acc = 0.0f;
#pragma unroll
    for (int h = 0; h < N_HEADS; ++h) {
        const float e = __expf(hwts[h] - hm);
        hsum += e;
        acc  += e * hout[((size_t)(b * N_HEADS + h)) * D_FEAT + f];
    }
    const float gate = 1.0f / (1.0f + __expf(-gate_logit[0]));
    out[idx] = x[idx] + gate * (acc / hsum);
}

// ---------------------------------------------------------------------------
extern "C" void kernel_launch(void* const* d_in, const int* in_sizes, int n_in,
                              void* d_out, int out_size, void* d_ws, size_t ws_size,
                              hipStream_t stream) {
    const float* x    = (const float*)d_in[0];   // [64,256]
    const float* Qe   = (const float*)d_in[1];   // [32,256,4]
    const float* Ke   = (const float*)d_in[2];   // [32,256,4]
    const float* Ve   = (const float*)d_in[3];   // [32,256,4]
    const float* op   = (const float*)d_in[4];   // [32,4]
    const float* hwts = (const float*)d_in[5];   // [32]
    const float* gl   = (const float*)d_in[6];   // [1]
    float* out = (float*)d_out;                  // [64,256]

    // Workspace layout
    char* ws = (char*)d_ws;
    float* G    = (float*)ws;                                        // 8 MB
    float* w    = (float*)(ws + (size_t)N_HEADS * D_FEAT * D_FEAT * sizeof(float));
    float* hout = w + (size_t)N_HEADS * D_FEAT;                      // 2 MB

    // 1) Gram tensor: 512 waves, 8 waves/block -> 64 blocks
    mha_gram_wmma_kernel<<<64, 256, 0, stream>>>(Qe, Ke, G);
    // 2) V projection: 32*256 = 8192 threads
    mha_vproj_kernel<<<N_HEADS, 256, 0, stream>>>(Ve, op, w);
    // 3) Fused softmax rows: one block per (b,h)
    mha_rows_kernel<<<BATCH_N * N_HEADS, 256, 0, stream>>>(x, G, w, hout);
    // 4) Combine heads + gated residual: 64*256 threads
    mha_combine_kernel<<<BATCH_N, 256, 0, stream>>>(x, hout, hwts, gl, out);
}